// NeighbourCovariance_36094905155927
// MI455X (gfx1250) — compile-verified
//
#include <hip/hip_runtime.h>
#include <hip/hip_bf16.h>

typedef __attribute__((ext_vector_type(2))) float v2f;
typedef __attribute__((ext_vector_type(8))) float v8f;

#define KNN 40
#define NF 32
#define WPB 8   // waves (vertices) per block

// One wave32 per vertex.
// GEMM per vertex: D[f,n] = sum_k A[k,f] * B[k,n]
//   A = w[k]*features[idx[k]][f]        (40 x 32)
//   B = [x,y,z, xx,xy,xz,yx,yy,yz,zx,zy,zz, 1, 0,0,0]  (40 x 16)
// via V_WMMA_F32_16X16X4_F32: 2 M-tiles x 10 k-steps = 20 WMMA / vertex.
__global__ __launch_bounds__(256) void neighbour_cov_wmma(
    const float* __restrict__ coords,    // V x 3
    const float* __restrict__ distsq,    // V x K
    const float* __restrict__ features,  // V x F
    const int*   __restrict__ n_idxs,    // V x K
    float* __restrict__ out,             // V x 384  (F*9 cov, F*3 means)
    int V)
{
    __shared__ float sC4[WPB][KNN * 4];               // {x,y,z,1} per neighbour
    __shared__ float sW [WPB][KNN];
    __shared__ int   sI [WPB][KNN];
    __shared__ __align__(16) float stage[WPB][384];   // output staging

    const int lane = threadIdx.x & 31;
    const int wv   = threadIdx.x >> 5;
    const int v    = blockIdx.x * WPB + wv;
    if (v >= V) return;

    const int lo   = lane & 15;   // N column
    const int hi   = lane >> 4;   // k sub-slot / row-half selector
    const int base = lane & 16;   // shuffle base for this half-wave

    // B-column selectors: col = c4[pa]*c4[pb]*nmask with c4 = {x,y,z,1}
    int pa, pb;
    if (lo < 3)       { pa = lo;             pb = 3; }
    else if (lo < 12) { pa = (lo - 3) / 3;   pb = (lo - 3) % 3; }
    else              { pa = 3;              pb = 3; }
    const float nmask = (lo <= 12) ? 1.0f : 0.0f;
    const int ci = (lo >= 3 && lo < 12) ? (lo - 3) / 3 : 0;
    const int di = (lo >= 3 && lo < 12) ? (lo - 3) % 3 : 0;

    // ---- Phase 1: stage weights / indices / neighbour coords into LDS ----
    for (int k = lane; k < KNN; k += 32) {
        int  raw   = n_idxs[(size_t)v * KNN + k];
        bool valid = (raw >= 0);
        int  idx   = valid ? raw : 0;
        float d    = distsq[(size_t)v * KNN + k];
        float w    = valid ? __expf(-10.0f * d) : 0.0f;  // invalid row => zero contribution
        sW[wv][k] = w;
        sI[wv][k] = idx;
        const float* cp = coords + (size_t)idx * 3;
        sC4[wv][k * 4 + 0] = cp[0];
        sC4[wv][k * 4 + 1] = cp[1];
        sC4[wv][k * 4 + 2] = cp[2];
        sC4[wv][k * 4 + 3] = 1.0f;
    }
    asm volatile("s_wait_dscnt 0" ::: "memory");  // intra-wave LDS RAW fence

    // ---- Phase 2: 20x V_WMMA_F32_16X16X4_F32 ----
    v8f acc0 = {0,0,0,0,0,0,0,0};   // features 0..15
    v8f acc1 = {0,0,0,0,0,0,0,0};   // features 16..31
#pragma unroll
    for (int s = 0; s < KNN / 4; ++s) {
        const int k0 = 4 * s + 2 * hi;    // A/B VGPR0 holds K=0 (lanes 0-15) / K=2 (lanes 16-31)
        const int k1 = k0 + 1;            // VGPR1 holds K=1 / K=3
        v2f B;
        B[0] = sC4[wv][k0 * 4 + pa] * sC4[wv][k0 * 4 + pb] * nmask;
        B[1] = sC4[wv][k1 * 4 + pa] * sC4[wv][k1 * 4 + pb] * nmask;
        const int   i0 = sI[wv][k0], i1 = sI[wv][k1];
        const float w0 = sW[wv][k0], w1 = sW[wv][k1];
        const float* f0 = features + (size_t)i0 * NF;
        const float* f1 = features + (size_t)i1 * NF;
        v2f A0, A1;
        A0[0] = w0 * f0[lo];       A0[1] = w1 * f1[lo];        // 64B coalesced per half-wave
        A1[0] = w0 * f0[16 + lo];  A1[1] = w1 * f1[16 + lo];
        acc0 = __builtin_amdgcn_wmma_f32_16x16x4_f32(
            false, A0, false, B, (short)0, acc0, false, false);
        acc1 = __builtin_amdgcn_wmma_f32_16x16x4_f32(
            false, A1, false, B, (short)0, acc1, false, false);
    }

    // ---- Phase 3: normalize, center covariance, stage output ----
#pragma unroll
    for (int j = 0; j < 8; ++j) {
        {   // M-tile 0: f = j + 8*hi
            float raw  = acc0[j];
            float den  = __shfl(raw, base + 12, 32) + 1e-3f;   // wsum[f] + EPS
            float valn = raw / den;
            float mc   = __shfl(valn, base + ci, 32);
            float md   = __shfl(valn, base + di, 32);
            int f = j + 8 * hi;
            if (lo < 3)       stage[wv][288 + f * 3 + lo] = valn;           // mean
            else if (lo < 12) stage[wv][f * 9 + (lo - 3)] = valn - mc * md; // cov
        }
        {   // M-tile 1: f = 16 + j + 8*hi
            float raw  = acc1[j];
            float den  = __shfl(raw, base + 12, 32) + 1e-3f;
            float valn = raw / den;
            float mc   = __shfl(valn, base + ci, 32);
            float md   = __shfl(valn, base + di, 32);
            int f = 16 + j + 8 * hi;
            if (lo < 3)       stage[wv][288 + f * 3 + lo] = valn;
            else if (lo < 12) stage[wv][f * 9 + (lo - 3)] = valn - mc * md;
        }
    }
    asm volatile("s_wait_dscnt 0" ::: "memory");

    // ---- Phase 4: coalesced 16B/lane output stores (384 floats = 3 x 32 x float4) ----
    const float4* s4 = (const float4*)&stage[wv][0];
    float4*       o4 = (float4*)(out + (size_t)v * 384);
#pragma unroll
    for (int t = 0; t < 3; ++t)
        o4[t * 32 + lane] = s4[t * 32 + lane];
}

extern "C" void kernel_launch(void* const* d_in, const int* in_sizes, int n_in,
                              void* d_out, int out_size, void* d_ws, size_t ws_size,
                              hipStream_t stream) {
    const float* coords   = (const float*)d_in[0];
    const float* distsq   = (const float*)d_in[1];
    const float* features = (const float*)d_in[2];
    const int*   n_idxs   = (const int*)d_in[3];
    float* out = (float*)d_out;

    const int V = in_sizes[0] / 3;                 // coordinates is V x 3
    const int blocks = (V + WPB - 1) / WPB;
    hipLaunchKernelGGL(neighbour_cov_wmma, dim3(blocks), dim3(256), 0, stream,
                       coords, distsq, features, n_idxs, out, V);
}